// AttentiveModes_22110491640334
// MI455X (gfx1250) — compile-verified
//
#include <hip/hip_runtime.h>
#include <math.h>

// ---------------------------------------------------------------------------
// AttentiveModes on MI455X (gfx1250), fp32 via V_WMMA_F32_16X16X4_F32.
// Memory-bound (~2 GB weight traffic over 3 rounds, 23.3 TB/s HBM => ~88us
// floor). Each GEMM wave computes a full M=32 x N=16 strip with two
// accumulators so every weight (B) fragment is loaded once and used twice:
// 3 b64 loads per 2 WMMAs in the steady-state loop.
// ---------------------------------------------------------------------------

namespace {
constexpr int S16 = 16;     // S
constexpr int C   = 1024;   // channels
constexpr int L   = 32;     // 2S rows
constexpr int D   = 512;    // H*DH
constexpr int NH  = 16;     // heads
constexpr int DH  = 32;     // head dim
constexpr int F   = 4096;   // W*C
constexpr float EPSV   = 1e-5f;
constexpr float QSCALE = 0.17677669529663687f;  // 1/sqrt(32)

// workspace offsets (in floats)
constexpr size_t OFF_XN  = 0;
constexpr size_t OFF_YN  = OFF_XN  + (size_t)S16 * L * C;   // 524288 each
constexpr size_t OFF_Q   = OFF_YN  + (size_t)S16 * L * C;
constexpr size_t OFF_K   = OFF_Q   + (size_t)S16 * L * D;   // 262144 each
constexpr size_t OFF_V   = OFF_K   + (size_t)S16 * L * D;
constexpr size_t OFF_O   = OFF_V   + (size_t)S16 * L * D;
constexpr size_t OFF_A   = OFF_O   + (size_t)S16 * L * D;
constexpr size_t OFF_X1  = OFF_A   + (size_t)S16 * L * C;
constexpr size_t OFF_X1N = OFF_X1  + (size_t)S16 * L * C;
constexpr size_t OFF_H   = OFF_X1N + (size_t)S16 * L * C;   // 16*32*4096
}  // namespace

typedef float v2f __attribute__((ext_vector_type(2)));
typedef float v8f __attribute__((ext_vector_type(8)));

__device__ __forceinline__ v8f wmma4(v2f a, v2f b, v8f c) {
  // D = A(16x4, f32) * B(4x16, f32) + C(16x16, f32)
  return __builtin_amdgcn_wmma_f32_16x16x4_f32(false, a, false, b, (short)0, c,
                                               false, false);
}

__device__ __forceinline__ float gelu_exact(float x) {
  return 0.5f * x * (1.0f + erff(x * 0.7071067811865475f));
}

// --- Round phase 1: gather a = [x[m1][i]; x[m2][:,i]^T], LN stats, xn/yn ----
__global__ __launch_bounds__(256) void k_assemble(
    const float* __restrict__ X, const float* __restrict__ lxw,
    const float* __restrict__ lxb, const float* __restrict__ lyw,
    const float* __restrict__ lyb, float* __restrict__ ws, int m1, int m2) {
  const int i = blockIdx.x;
  const int tid = threadIdx.x;
  float* aB = ws + OFF_A + (size_t)i * L * C;
  float* xn = ws + OFF_XN + (size_t)i * L * C;
  float* yn = ws + OFF_YN + (size_t)i * L * C;

  float s = 0.f, s2 = 0.f;
  for (int e = tid; e < L * C; e += 256) {
    const int l = e >> 10, c = e & (C - 1);
    const float v =
        (l < S16) ? X[(((size_t)m1 * S16 + i) * S16 + l) * C + c]
                  : X[(((size_t)m2 * S16 + (l - S16)) * S16 + i) * C + c];
    aB[e] = v;
    s += v;
    s2 += v * v;
  }
  __shared__ float r0[8], r1[8], mv[2];
  for (int o = 16; o > 0; o >>= 1) {
    s += __shfl_down(s, o, 32);
    s2 += __shfl_down(s2, o, 32);
  }
  if ((tid & 31) == 0) { r0[tid >> 5] = s; r1[tid >> 5] = s2; }
  __syncthreads();
  if (tid < 32) {
    s = (tid < 8) ? r0[tid] : 0.f;
    s2 = (tid < 8) ? r1[tid] : 0.f;
    for (int o = 4; o > 0; o >>= 1) {
      s += __shfl_down(s, o, 32);
      s2 += __shfl_down(s2, o, 32);
    }
    if (tid == 0) {
      const float m = s / (float)(L * C);
      const float var = fmaxf(s2 / (float)(L * C) - m * m, 0.f);
      mv[0] = m;
      mv[1] = rsqrtf(var + EPSV);
    }
  }
  __syncthreads();
  const float m = mv[0], inv = mv[1];
  const size_t pb = (size_t)i * L * C;
  for (int e = tid; e < L * C; e += 256) {
    const float t = (aB[e] - m) * inv;
    xn[e] = t * lxw[pb + e] + lxb[pb + e];
    yn[e] = t * lyw[pb + e] + lyb[pb + e];
  }
}

// --- Phase 2: Q/K/V GEMMs, [32,1024]x[1024,512]; M=32 strip per wave -------
__global__ __launch_bounds__(256) void k_qkv(
    float* __restrict__ ws, const float* __restrict__ wq,
    const float* __restrict__ bq, const float* __restrict__ wk,
    const float* __restrict__ bk, const float* __restrict__ wv,
    const float* __restrict__ bv) {
  const int wave = blockIdx.x * 8 + (threadIdx.x >> 5);
  const int lane = threadIdx.x & 31;
  constexpr int NT = D / 16;           // 32
  constexpr int total = S16 * 3 * NT;  // 1536
  if (wave >= total) return;
  int t = wave;
  const int i = t / (3 * NT); t -= i * 3 * NT;
  const int mat = t / NT;
  const int nt = t - mat * NT;

  const float* A = ws + (mat == 0 ? OFF_XN : OFF_YN) + (size_t)i * L * C;
  const float* Wm = (mat == 0 ? wq : (mat == 1 ? wk : wv)) + (size_t)i * D * C;
  const float* bs = (mat == 0 ? bq : (mat == 1 ? bk : bv)) + (size_t)i * D;
  float* Out = ws + (mat == 0 ? OFF_Q : (mat == 1 ? OFF_K : OFF_V)) +
               (size_t)i * L * D;

  const int hl = lane >> 4, l16 = lane & 15;
  const int n0 = nt * 16;
  const float* aP0 = A + (size_t)l16 * C + 2 * hl;         // rows 0..15
  const float* aP1 = A + (size_t)(16 + l16) * C + 2 * hl;  // rows 16..31
  const float* bP = Wm + (size_t)(n0 + l16) * C + 2 * hl;  // B[k][n]=w[n][k]
  v8f acc0 = {}, acc1 = {};
#pragma unroll 8
  for (int k0 = 0; k0 < C; k0 += 4) {
    const v2f b = *(const v2f*)(bP + k0);
    const v2f a0 = *(const v2f*)(aP0 + k0);
    const v2f a1 = *(const v2f*)(aP1 + k0);
    acc0 = wmma4(a0, b, acc0);
    acc1 = wmma4(a1, b, acc1);
  }
  const float bc = bs[n0 + l16];
#pragma unroll
  for (int j = 0; j < 8; ++j) {
    Out[(size_t)(8 * hl + j) * D + n0 + l16] = acc0[j] + bc;
    Out[(size_t)(16 + 8 * hl + j) * D + n0 + l16] = acc1[j] + bc;
  }
}

// --- Phase 3: per (idx, head): scores = q k^T * scale; o = scores v --------
__global__ __launch_bounds__(128) void k_attn(float* __restrict__ ws) {
  __shared__ float sc[32][33];
  const int i = blockIdx.x >> 4;
  const int h = blockIdx.x & 15;
  const int wv_ = threadIdx.x >> 5;  // 4 waves: 2x2 tiles
  const int lane = threadIdx.x & 31;
  const int hl = lane >> 4, l16 = lane & 15;
  const float* Q = ws + OFF_Q + (size_t)i * L * D + h * DH;
  const float* Kp = ws + OFF_K + (size_t)i * L * D + h * DH;
  const float* Vp = ws + OFF_V + (size_t)i * L * D + h * DH;
  float* O = ws + OFF_O + (size_t)i * L * D + h * DH;
  const int m0 = (wv_ >> 1) * 16, n0 = (wv_ & 1) * 16;
  {
    // scores tile: A = q (K=DH), B[k][n] = k[n][k]
    const float* aP = Q + (size_t)(m0 + l16) * D + 2 * hl;
    const float* bP = Kp + (size_t)(n0 + l16) * D + 2 * hl;
    v8f acc = {};
#pragma unroll
    for (int k0 = 0; k0 < DH; k0 += 4) {
      const v2f a = *(const v2f*)(aP + k0);
      const v2f b = *(const v2f*)(bP + k0);
      acc = wmma4(a, b, acc);
    }
#pragma unroll
    for (int j = 0; j < 8; ++j)
      sc[m0 + 8 * hl + j][n0 + l16] = acc[j] * QSCALE;
  }
  __syncthreads();
  {
    // o tile: A = scores (K=32 key rows), B[k][n] = v[k][n]
    v8f acc = {};
#pragma unroll
    for (int k0 = 0; k0 < L; k0 += 4) {
      v2f a, b;
      a.x = sc[m0 + l16][k0 + 2 * hl];
      a.y = sc[m0 + l16][k0 + 2 * hl + 1];
      b.x = Vp[(size_t)(k0 + 2 * hl) * D + n0 + l16];
      b.y = Vp[(size_t)(k0 + 2 * hl + 1) * D + n0 + l16];
      acc = wmma4(a, b, acc);
    }
#pragma unroll
    for (int j = 0; j < 8; ++j)
      O[(size_t)(m0 + 8 * hl + j) * D + n0 + l16] = acc[j];
  }
}

// --- Phase 4: x1 = a + o @ w1^T + b1, [32,512]x[512,1024] ------------------
__global__ __launch_bounds__(256) void k_x1(float* __restrict__ ws,
                                            const float* __restrict__ w1,
                                            const float* __restrict__ b1) {
  const int wave = blockIdx.x * 8 + (threadIdx.x >> 5);
  const int lane = threadIdx.x & 31;
  constexpr int NT = C / 16;       // 64
  constexpr int total = S16 * NT;  // 1024
  if (wave >= total) return;
  const int i = wave / NT;
  const int nt = wave - i * NT;
  const float* A = ws + OFF_O + (size_t)i * L * D;
  const float* Wm = w1 + (size_t)i * C * D;
  const float* aB = ws + OFF_A + (size_t)i * L * C;
  float* X1 = ws + OFF_X1 + (size_t)i * L * C;
  const int hl = lane >> 4, l16 = lane & 15;
  const int n0 = nt * 16;
  const float* aP0 = A + (size_t)l16 * D + 2 * hl;
  const float* aP1 = A + (size_t)(16 + l16) * D + 2 * hl;
  const float* bP = Wm + (size_t)(n0 + l16) * D + 2 * hl;
  v8f acc0 = {}, acc1 = {};
#pragma unroll 8
  for (int k0 = 0; k0 < D; k0 += 4) {
    const v2f b = *(const v2f*)(bP + k0);
    const v2f a0 = *(const v2f*)(aP0 + k0);
    const v2f a1 = *(const v2f*)(aP1 + k0);
    acc0 = wmma4(a0, b, acc0);
    acc1 = wmma4(a1, b, acc1);
  }
  const float bc = b1[(size_t)i * C + n0 + l16];
#pragma unroll
  for (int j = 0; j < 8; ++j) {
    const size_t i0 = (size_t)(8 * hl + j) * C + n0 + l16;
    const size_t i1 = (size_t)(16 + 8 * hl + j) * C + n0 + l16;
    X1[i0] = acc0[j] + bc + aB[i0];
    X1[i1] = acc1[j] + bc + aB[i1];
  }
}

// --- Phase 5: LN(x1) -> x1n -------------------------------------------------
__global__ __launch_bounds__(256) void k_stats2(float* __restrict__ ws,
                                                const float* __restrict__ lfw,
                                                const float* __restrict__ lfb) {
  const int i = blockIdx.x;
  const int tid = threadIdx.x;
  const float* X1 = ws + OFF_X1 + (size_t)i * L * C;
  float* X1N = ws + OFF_X1N + (size_t)i * L * C;
  float s = 0.f, s2 = 0.f;
  for (int e = tid; e < L * C; e += 256) {
    const float v = X1[e];
    s += v;
    s2 += v * v;
  }
  __shared__ float r0[8], r1[8], mv[2];
  for (int o = 16; o > 0; o >>= 1) {
    s += __shfl_down(s, o, 32);
    s2 += __shfl_down(s2, o, 32);
  }
  if ((tid & 31) == 0) { r0[tid >> 5] = s; r1[tid >> 5] = s2; }
  __syncthreads();
  if (tid < 32) {
    s = (tid < 8) ? r0[tid] : 0.f;
    s2 = (tid < 8) ? r1[tid] : 0.f;
    for (int o = 4; o > 0; o >>= 1) {
      s += __shfl_down(s, o, 32);
      s2 += __shfl_down(s2, o, 32);
    }
    if (tid == 0) {
      const float m = s / (float)(L * C);
      const float var = fmaxf(s2 / (float)(L * C) - m * m, 0.f);
      mv[0] = m;
      mv[1] = rsqrtf(var + EPSV);
    }
  }
  __syncthreads();
  const float m = mv[0], inv = mv[1];
  const size_t pb = (size_t)i * L * C;
  for (int e = tid; e < L * C; e += 256) {
    const float t = (X1[e] - m) * inv;
    X1N[e] = t * lfw[pb + e] + lfb[pb + e];
  }
}

// --- Phase 6: h = gelu(x1n @ w2^T + b2), [32,1024]x[1024,4096] -------------
__global__ __launch_bounds__(256) void k_h(float* __restrict__ ws,
                                           const float* __restrict__ w2,
                                           const float* __restrict__ b2) {
  const int wave = blockIdx.x * 8 + (threadIdx.x >> 5);
  const int lane = threadIdx.x & 31;
  constexpr int NT = F / 16;       // 256
  constexpr int total = S16 * NT;  // 4096
  if (wave >= total) return;
  const int i = wave / NT;
  const int nt = wave - i * NT;
  const float* A = ws + OFF_X1N + (size_t)i * L * C;
  const float* Wm = w2 + (size_t)i * F * C;
  float* Hb = ws + OFF_H + (size_t)i * L * F;
  const int hl = lane >> 4, l16 = lane & 15;
  const int n0 = nt * 16;
  const float* aP0 = A + (size_t)l16 * C + 2 * hl;
  const float* aP1 = A + (size_t)(16 + l16) * C + 2 * hl;
  const float* bP = Wm + (size_t)(n0 + l16) * C + 2 * hl;
  v8f acc0 = {}, acc1 = {};
#pragma unroll 8
  for (int k0 = 0; k0 < C; k0 += 4) {
    const v2f b = *(const v2f*)(bP + k0);
    const v2f a0 = *(const v2f*)(aP0 + k0);
    const v2f a1 = *(const v2f*)(aP1 + k0);
    acc0 = wmma4(a0, b, acc0);
    acc1 = wmma4(a1, b, acc1);
  }
  const float bc = b2[(size_t)i * F + n0 + l16];
#pragma unroll
  for (int j = 0; j < 8; ++j) {
    Hb[(size_t)(8 * hl + j) * F + n0 + l16] = gelu_exact(acc0[j] + bc);
    Hb[(size_t)(16 + 8 * hl + j) * F + n0 + l16] = gelu_exact(acc1[j] + bc);
  }
}

// --- Phase 7: out = x1 + h @ w3^T + b3, scatter into x ---------------------
__global__ __launch_bounds__(256) void k_out(const float* __restrict__ ws,
                                             const float* __restrict__ w3,
                                             const float* __restrict__ b3,
                                             float* __restrict__ X, int m1,
                                             int m2) {
  const int wave = blockIdx.x * 8 + (threadIdx.x >> 5);
  const int lane = threadIdx.x & 31;
  constexpr int NT = C / 16;       // 64
  constexpr int total = S16 * NT;  // 1024
  if (wave >= total) return;
  const int i = wave / NT;
  const int nt = wave - i * NT;
  const float* A = ws + OFF_H + (size_t)i * L * F;
  const float* Wm = w3 + (size_t)i * C * F;
  const float* X1 = ws + OFF_X1 + (size_t)i * L * C;
  const int hl = lane >> 4, l16 = lane & 15;
  const int n0 = nt * 16;
  const float* aP0 = A + (size_t)l16 * F + 2 * hl;
  const float* aP1 = A + (size_t)(16 + l16) * F + 2 * hl;
  const float* bP = Wm + (size_t)(n0 + l16) * F + 2 * hl;
  v8f acc0 = {}, acc1 = {};
#pragma unroll 8
  for (int k0 = 0; k0 < F; k0 += 4) {
    const v2f b = *(const v2f*)(bP + k0);
    const v2f a0 = *(const v2f*)(aP0 + k0);
    const v2f a1 = *(const v2f*)(aP1 + k0);
    acc0 = wmma4(a0, b, acc0);
    acc1 = wmma4(a1, b, acc1);
  }
  const int cc = n0 + l16;
  const float bc = b3[(size_t)i * C + cc];
#pragma unroll
  for (int j = 0; j < 8; ++j) {
    // rows 0..15 -> x[m1][i][l]
    {
      const int l = 8 * hl + j;
      const float val = acc0[j] + bc + X1[(size_t)l * C + cc];
      X[(((size_t)m1 * S16 + i) * S16 + l) * C + cc] = val;
    }
    // rows 16..31 -> x[m2][i][l-16] (transpose NOT undone — faithful to ref)
    {
      const int l = 16 + 8 * hl + j;
      const float val = acc1[j] + bc + X1[(size_t)l * C + cc];
      X[(((size_t)m2 * S16 + i) * S16 + (l - S16)) * C + cc] = val;
    }
  }
}

extern "C" void kernel_launch(void* const* d_in, const int* in_sizes, int n_in,
                              void* d_out, int out_size, void* d_ws,
                              size_t ws_size, hipStream_t stream) {
  const float* x   = (const float*)d_in[0];
  const float* lxw = (const float*)d_in[1];
  const float* lxb = (const float*)d_in[2];
  const float* lyw = (const float*)d_in[3];
  const float* lyb = (const float*)d_in[4];
  const float* lfw = (const float*)d_in[5];
  const float* lfb = (const float*)d_in[6];
  const float* wq  = (const float*)d_in[7];
  const float* bq  = (const float*)d_in[8];
  const float* wk  = (const float*)d_in[9];
  const float* bk  = (const float*)d_in[10];
  const float* wv  = (const float*)d_in[11];
  const float* bv  = (const float*)d_in[12];
  const float* w1  = (const float*)d_in[13];
  const float* b1  = (const float*)d_in[14];
  const float* w2  = (const float*)d_in[15];
  const float* b2  = (const float*)d_in[16];
  const float* w3  = (const float*)d_in[17];
  const float* b3  = (const float*)d_in[18];
  float* X = (float*)d_out;
  float* ws = (float*)d_ws;

  hipMemcpyAsync(X, x, (size_t)3 * S16 * S16 * C * sizeof(float),
                 hipMemcpyDeviceToDevice, stream);

  const int rounds[3][2] = {{0, 1}, {2, 0}, {1, 2}};
  for (int r = 0; r < 3; ++r) {
    const int m1 = rounds[r][0], m2 = rounds[r][1];
    k_assemble<<<16, 256, 0, stream>>>(X, lxw, lxb, lyw, lyb, ws, m1, m2);
    k_qkv<<<192, 256, 0, stream>>>(ws, wq, bq, wk, bk, wv, bv);
    k_attn<<<256, 128, 0, stream>>>(ws);
    k_x1<<<128, 256, 0, stream>>>(ws, w1, b1);
    k_stats2<<<16, 256, 0, stream>>>(ws, lfw, lfb);
    k_h<<<512, 256, 0, stream>>>(ws, w2, b2);
    k_out<<<128, 256, 0, stream>>>(ws, w3, b3, X, m1, m2);
  }
}